// CumulativeSetFeatures_38508676776565
// MI455X (gfx1250) — compile-verified
//
#include <hip/hip_runtime.h>

// Problem constants (from reference)
#define B_  16
#define C_  256
#define L_  4096
#define P_  100
#define Q_  20
#define MT_ 7          // ceil(P/16) m-tiles
#define NL  4          // l-tiles (16 cols each) per wave
#define PQ_ (P_ * Q_)

typedef __attribute__((ext_vector_type(2))) float v2f;
typedef __attribute__((ext_vector_type(8))) float v8f;

__global__ void csf_zero_counts(int* __restrict__ counts) {
    int i = blockIdx.x * 256 + threadIdx.x;
    if (i < B_ * PQ_) counts[i] = 0;
}

// One wave per (b, mtile, group of NL ltiles).
// GEMM tile via V_WMMA_F32_16X16X4_F32, fused threshold + NT set_out stores +
// ballot-popcount count accumulation (integer atomics -> deterministic).
__global__ __launch_bounds__(32)
void csf_main(const float* __restrict__ X,       // [B,C,L]
              const float* __restrict__ proj,    // [P,C]
              const float* __restrict__ minv,    // [P]
              const float* __restrict__ maxv,    // [P]
              float* __restrict__ set_out,       // [B, P*Q, L]
              int* __restrict__ counts) {        // [B, P, Q]
    const int lane  = threadIdx.x;       // 0..31
    const int b     = blockIdx.z;
    const int mt    = blockIdx.y;        // 0..6
    const int lt0   = blockIdx.x * NL;   // base l-tile
    const int lsub  = lane & 15;
    const int khalf = lane >> 4;

    // ---- A-matrix (16x4 f32): lanes 0-15 hold M=0..15 K={0,1}; lanes 16-31 K={2,3}
    int mrow   = mt * 16 + lsub;
    int mclamp = mrow < P_ ? mrow : (P_ - 1);     // clamp so EXEC stays full for WMMA
    const float* Arow = proj + (size_t)mclamp * C_ + 2 * khalf;

    // ---- B-matrix (4x16 f32): lanes 0-15 N=0..15 K={0,1}; lanes 16-31 K={2,3}
    const float* Xb = X + (size_t)b * C_ * L_;

    v8f c[NL];
#pragma unroll
    for (int t = 0; t < NL; ++t) c[t] = (v8f){0.f,0.f,0.f,0.f,0.f,0.f,0.f,0.f};

    for (int k0 = 0; k0 < C_; k0 += 4) {
        v2f a = *(const v2f*)(Arow + k0);         // K = k0+2*khalf, k0+2*khalf+1
        const float* Bk = Xb + (size_t)(k0 + 2 * khalf) * L_ + lt0 * 16 + lsub;
#pragma unroll
        for (int t = 0; t < NL; ++t) {
            v2f bb;
            bb.x = Bk[t * 16];        // K = k0+2*khalf
            bb.y = Bk[t * 16 + L_];   // K = k0+2*khalf+1
            c[t] = __builtin_amdgcn_wmma_f32_16x16x4_f32(
                false, a, false, bb, (short)0, c[t], false, false);
        }
    }

    // ---- Epilogue: thresholds, set_out stores, ballot counts ----
    // C/D layout: VGPR i -> row (mt*16 + i) in lanes 0-15, row (mt*16+8+i) in lanes 16-31.
    int acc[Q_];
#pragma unroll
    for (int q = 0; q < Q_; ++q) acc[q] = 0;

    float* setb = set_out + (size_t)b * PQ_ * L_;
    const float invQ1 = 1.0f / (float)(Q_ + 1);

#pragma unroll
    for (int i = 0; i < 8; ++i) {
        int  p     = mt * 16 + i + 8 * khalf;   // row this lane's c[.][i] belongs to
        bool valid = p < P_;
        int  pc    = valid ? p : 0;
        float mn   = minv[pc];
        float mx   = maxv[pc];
        float step = (mx - mn) * invQ1;

        // Row-block base pointer: all (q,t) offsets below fit the 24-bit store imm.
        float* base_i = setb + (size_t)p * Q_ * L_ + (size_t)lt0 * 16 + lsub;

        // Lane-select mask for count accumulation; encodes row validity so the
        // ballot predicate can be the raw compare.
        bool row_lo_ok = (mt * 16 + i) < P_;
        bool row_hi_ok = (mt * 16 + 8 + i) < P_;
        unsigned msel = (lane == i)     ? (row_lo_ok ? 0xFFFFu : 0u)
                      : (lane == 8 + i) ? (row_hi_ok ? 0xFFFF0000u : 0u)
                                        : 0u;
#pragma unroll
        for (int q = 0; q < Q_; ++q) {
            float thr = mn + step * (float)(q + 1);
#pragma unroll
            for (int t = 0; t < NL; ++t) {
                bool s = c[t][i] < thr;
                unsigned mask = __builtin_amdgcn_ballot_w32(s);   // full-EXEC here
                if (valid)
                    __builtin_nontemporal_store(s ? 1.0f : 0.0f,
                                                base_i + q * L_ + t * 16);
                acc[q] += __builtin_popcount(mask & msel);
            }
        }
    }

    // lane s in 0..15 owns row mt*16+s; flush integer counts (exact -> deterministic)
    int prow = mt * 16 + lane;
    if (lane < 16 && prow < P_) {
        int base = (b * P_ + prow) * Q_;
#pragma unroll
        for (int q = 0; q < Q_; ++q)
            atomicAdd(&counts[base + q], acc[q]);
    }
}

__global__ void csf_finalize(const int* __restrict__ counts, float* __restrict__ cdf) {
    int i = blockIdx.x * 256 + threadIdx.x;
    if (i < B_ * PQ_) cdf[i] = (float)counts[i] * (1.0f / (float)L_);
}

extern "C" void kernel_launch(void* const* d_in, const int* in_sizes, int n_in,
                              void* d_out, int out_size, void* d_ws, size_t ws_size,
                              hipStream_t stream) {
    const float* X    = (const float*)d_in[0];   // [B,C,L]
    const float* proj = (const float*)d_in[1];   // [P,C]
    const float* minv = (const float*)d_in[2];   // [P]
    const float* maxv = (const float*)d_in[3];   // [P]

    float* cdf_out = (float*)d_out;              // [B, P*Q]
    float* set_out = cdf_out + (size_t)B_ * PQ_; // [B, P*Q, L]
    int*   counts  = (int*)d_ws;                 // 128 KB scratch

    csf_zero_counts<<<(B_ * PQ_ + 255) / 256, 256, 0, stream>>>(counts);

    dim3 grid(L_ / (16 * NL), MT_, B_);          // 64 x 7 x 16 waves
    csf_main<<<grid, 32, 0, stream>>>(X, proj, minv, maxv, set_out, counts);

    csf_finalize<<<(B_ * PQ_ + 255) / 256, 256, 0, stream>>>(counts, cdf_out);
}